// CMHSA_37907381354671
// MI455X (gfx1250) — compile-verified
//
#include <hip/hip_runtime.h>
#include <hip/hip_bf16.h>

#define Bn 4
#define Cc 512
#define Tt 1024
#define HEADS 8
#define HD 64
#define EPS 1e-5f
#define NB 4   // N-tiles (of 16) per wave: 16x64 output tile, A reused 4x

typedef float v2f __attribute__((ext_vector_type(2)));
typedef float v8f __attribute__((ext_vector_type(8)));

// D = A(16x4) * B(4x16) + C, fp32 WMMA (CDNA5 V_WMMA_F32_16X16X4_F32)
__device__ __forceinline__ v8f wmma4(v2f a, v2f b, v8f c) {
  return __builtin_amdgcn_wmma_f32_16x16x4_f32(
      /*neg_a=*/false, a, /*neg_b=*/false, b,
      /*c_mod=*/(short)0, c, /*reuse_a=*/false, /*reuse_b=*/false);
}

// ---------------------------------------------------------------------------
// Kernel 1: Q/K/V 1x1 convs.  O[b,which,o,t] = sum_c W[o,c] * X[b,c,t]
// grid = (T/64, C/16, B*3), block = 32; one wave -> 16x64 tile
// ---------------------------------------------------------------------------
__global__ void qkv_kernel(const float* __restrict__ x,
                           const float* __restrict__ wq,
                           const float* __restrict__ wk,
                           const float* __restrict__ wv,
                           float* __restrict__ qkv) {
  const int lane = threadIdx.x & 31;
  const int half = lane >> 4;
  const int l16  = lane & 15;
  const int nbase = blockIdx.x * (16 * NB); // t
  const int mbase = blockIdx.y * 16;        // out channel
  const int b     = blockIdx.z / 3;
  const int which = blockIdx.z % 3;
  const float* W = (which == 0) ? wq : (which == 1 ? wk : wv);
  const float* X = x + (size_t)b * Cc * Tt;
  const float* wrow = W + (size_t)(mbase + l16) * Cc;

  v8f acc[NB] = {};
#pragma unroll 4
  for (int k = 0; k < Cc; k += 4) {
    const int ka = k + half * 2;
    const v2f a = *(const v2f*)(wrow + ka);            // contiguous pair
    const float* x0 = X + (size_t)ka * Tt + nbase + l16;
    const float* x1 = x0 + Tt;
#pragma unroll
    for (int j = 0; j < NB; ++j) {
      v2f bb; bb.x = x0[j * 16]; bb.y = x1[j * 16];
      acc[j] = wmma4(a, bb, acc[j]);
    }
  }

  float* O = qkv + ((size_t)b * 3 + which) * Cc * Tt;
#pragma unroll
  for (int j = 0; j < NB; ++j)
#pragma unroll
    for (int r = 0; r < 8; ++r)
      O[(size_t)(mbase + r + half * 8) * Tt + nbase + j * 16 + l16] = acc[j][r];
}

// ---------------------------------------------------------------------------
// Kernel 2: head-mixed scores.
// S[b,g,q,t] = sum_c (w_head[g][c/64]*scale) * Q[b,c,q] * K[b,c,t]
// grid = (T/64, T/16, B*HEADS), block = 32; one wave -> 16x64 tile
// ---------------------------------------------------------------------------
__global__ void score_kernel(const float* __restrict__ qkv,
                             const float* __restrict__ w_head,
                             float* __restrict__ S) {
  const int lane = threadIdx.x & 31;
  const int half = lane >> 4;
  const int l16  = lane & 15;
  const int nbase = blockIdx.x * (16 * NB); // t (key)
  const int mbase = blockIdx.y * 16;        // q (query)
  const int bh = blockIdx.z;
  const int b = bh / HEADS;
  const int g = bh % HEADS;
  const float* Q = qkv + ((size_t)b * 3 + 0) * Cc * Tt;
  const float* K = qkv + ((size_t)b * 3 + 1) * Cc * Tt;

  const float scale = 0.125f;  // hd^-0.5
  float wh[HEADS];
#pragma unroll
  for (int h = 0; h < HEADS; ++h) wh[h] = w_head[g * HEADS + h] * scale;

  const int qcol = mbase + l16;
  v8f acc[NB] = {};
#pragma unroll 4
  for (int k = 0; k < Cc; k += 4) {
    const int ka = k + half * 2;            // ka, ka+1 in same 64-block
    const float s = wh[ka >> 6];
    v2f a;  a.x = Q[(size_t)ka * Tt + qcol] * s;       // gather, loaded ONCE
            a.y = Q[(size_t)(ka + 1) * Tt + qcol] * s; // reused across NB WMMAs
    const float* k0 = K + (size_t)ka * Tt + nbase + l16;
    const float* k1 = k0 + Tt;
#pragma unroll
    for (int j = 0; j < NB; ++j) {
      v2f bb; bb.x = k0[j * 16]; bb.y = k1[j * 16];
      acc[j] = wmma4(a, bb, acc[j]);
    }
  }

  float* Sp = S + (size_t)bh * Tt * Tt;
#pragma unroll
  for (int j = 0; j < NB; ++j)
#pragma unroll
    for (int r = 0; r < 8; ++r)
      Sp[(size_t)(mbase + r + half * 8) * Tt + nbase + j * 16 + l16] = acc[j][r];
}

// ---------------------------------------------------------------------------
// Kernel 3: in-place row softmax + per-(b,g) sumsq (InstanceNorm stats).
// Softmax rows sum to 1 => mean is analytically 1/T; only sumsq reduced.
// grid = (T, B*HEADS), block = 256
// ---------------------------------------------------------------------------
__global__ void softmax_kernel(float* __restrict__ S, float* __restrict__ sumsq) {
  __shared__ float red[256];
  const int tid = threadIdx.x;
  float* p = S + ((size_t)blockIdx.y * Tt + blockIdx.x) * Tt;

  float4 v = reinterpret_cast<float4*>(p)[tid];

  float m = fmaxf(fmaxf(v.x, v.y), fmaxf(v.z, v.w));
  red[tid] = m; __syncthreads();
  for (int s = 128; s > 0; s >>= 1) {
    if (tid < s) red[tid] = fmaxf(red[tid], red[tid + s]);
    __syncthreads();
  }
  m = red[0]; __syncthreads();

  v.x = expf(v.x - m); v.y = expf(v.y - m);
  v.z = expf(v.z - m); v.w = expf(v.w - m);

  red[tid] = v.x + v.y + v.z + v.w; __syncthreads();
  for (int s = 128; s > 0; s >>= 1) {
    if (tid < s) red[tid] += red[tid + s];
    __syncthreads();
  }
  const float inv = 1.0f / red[0]; __syncthreads();

  v.x *= inv; v.y *= inv; v.z *= inv; v.w *= inv;
  reinterpret_cast<float4*>(p)[tid] = v;

  red[tid] = v.x * v.x + v.y * v.y + v.z * v.z + v.w * v.w; __syncthreads();
  for (int s = 128; s > 0; s >>= 1) {
    if (tid < s) red[tid] += red[tid + s];
    __syncthreads();
  }
  if (tid == 0) atomicAdd(&sumsq[blockIdx.y], red[0]);
}

// ---------------------------------------------------------------------------
// Kernel 3b: vsum[b,c] = sum_t V[b,c,t]
// ---------------------------------------------------------------------------
__global__ void vsum_kernel(const float* __restrict__ qkv, float* __restrict__ vsum) {
  const int idx = blockIdx.x * blockDim.x + threadIdx.x;  // b*Cc + c
  const int b = idx / Cc, c = idx % Cc;
  const float* V = qkv + ((size_t)b * 3 + 2) * Cc * Tt + (size_t)c * Tt;
  float s = 0.f;
  for (int t = 0; t < Tt; ++t) s += V[t];
  vsum[idx] = s;
}

// ---------------------------------------------------------------------------
// Kernel 4: out = normalize(attn) @ V^T, affine norm folded:
//   out[q,d] = g*istd*(P@V^T)[q,d] + (beta - g*istd*mean)*vsum[d]
// One wave covers all hd=64 (NB n-tiles). Stores into .view-scrambled [B,T,C]:
//   row = h*128 + (q>>3), col = ((q&7)<<6) + d
// grid = (T/16, B*HEADS), block = 32
// ---------------------------------------------------------------------------
__global__ void av_kernel(const float* __restrict__ qkv,
                          const float* __restrict__ S,
                          const float* __restrict__ sumsq,
                          const float* __restrict__ vsum,
                          const float* __restrict__ in_gamma,
                          const float* __restrict__ in_beta,
                          float* __restrict__ out1) {
  const int lane = threadIdx.x & 31;
  const int half = lane >> 4;
  const int l16  = lane & 15;
  const int mbase = blockIdx.x * 16;        // q
  const int bh = blockIdx.y;
  const int b = bh / HEADS;
  const int h = bh % HEADS;
  const float* P = S + (size_t)bh * Tt * Tt;
  const float* V = qkv + ((size_t)b * 3 + 2) * Cc * Tt + (size_t)h * HD * Tt;

  const float* arow = P + (size_t)(mbase + l16) * Tt;
  const float* br[NB];
#pragma unroll
  for (int j = 0; j < NB; ++j) br[j] = V + (size_t)(j * 16 + l16) * Tt;

  v8f acc[NB] = {};
#pragma unroll 4
  for (int k = 0; k < Tt; k += 4) {
    const int ka = k + half * 2;
    const v2f a = *(const v2f*)(arow + ka);            // contiguous pair, reused
#pragma unroll
    for (int j = 0; j < NB; ++j) {
      const v2f bb = *(const v2f*)(br[j] + ka);        // contiguous pair
      acc[j] = wmma4(a, bb, acc[j]);
    }
  }

  const float mean = 1.0f / (float)Tt;
  const float var  = sumsq[bh] / ((float)Tt * (float)Tt) - mean * mean;
  const float istd = rsqrtf(var + EPS);
  const float gs   = in_gamma[h] * istd;
  const float cadd = in_beta[h] - gs * mean;

#pragma unroll
  for (int j = 0; j < NB; ++j) {
    const int d = j * 16 + l16;
    const float vs = cadd * vsum[b * Cc + h * HD + d];
#pragma unroll
    for (int r = 0; r < 8; ++r) {
      const int q = mbase + r + half * 8;
      const size_t row = (size_t)(h * 128 + (q >> 3));
      const size_t col = (size_t)(((q & 7) << 6) + d);
      out1[(size_t)b * Tt * Cc + row * Cc + col] = gs * acc[j][r] + vs;
    }
  }
}

// ---------------------------------------------------------------------------
// Kernel 5: projection. final[b,co,t] = sum_c out1[b,t,c]*w_proj[co,c] + b_proj[co]
// grid = (T/64, C/16, B), block = 32; one wave -> 16x64 tile
// ---------------------------------------------------------------------------
__global__ void proj_kernel(const float* __restrict__ out1,
                            const float* __restrict__ w_proj,
                            const float* __restrict__ b_proj,
                            float* __restrict__ out) {
  const int lane = threadIdx.x & 31;
  const int half = lane >> 4;
  const int l16  = lane & 15;
  const int nbase = blockIdx.x * (16 * NB); // t
  const int mbase = blockIdx.y * 16;        // out channel
  const int b = blockIdx.z;
  const float* wrow = w_proj + (size_t)(mbase + l16) * Cc;
  const float* br[NB];
#pragma unroll
  for (int j = 0; j < NB; ++j)
    br[j] = out1 + (size_t)b * Tt * Cc + (size_t)(nbase + j * 16 + l16) * Cc;

  v8f acc[NB] = {};
#pragma unroll 4
  for (int k = 0; k < Cc; k += 4) {
    const int ka = k + half * 2;
    const v2f a = *(const v2f*)(wrow + ka);            // contiguous pair, reused
#pragma unroll
    for (int j = 0; j < NB; ++j) {
      const v2f bb = *(const v2f*)(br[j] + ka);        // contiguous pair
      acc[j] = wmma4(a, bb, acc[j]);
    }
  }

#pragma unroll
  for (int j = 0; j < NB; ++j)
#pragma unroll
    for (int r = 0; r < 8; ++r) {
      const int row = mbase + r + half * 8;
      out[(size_t)b * Cc * Tt + (size_t)row * Tt + nbase + j * 16 + l16] =
          acc[j][r] + b_proj[row];
    }
}

extern "C" void kernel_launch(void* const* d_in, const int* in_sizes, int n_in,
                              void* d_out, int out_size, void* d_ws, size_t ws_size,
                              hipStream_t stream) {
  const float* x        = (const float*)d_in[0];
  const float* w_q      = (const float*)d_in[1];
  const float* w_k      = (const float*)d_in[2];
  const float* w_v      = (const float*)d_in[3];
  const float* w_head   = (const float*)d_in[4];
  const float* in_gamma = (const float*)d_in[5];
  const float* in_beta  = (const float*)d_in[6];
  const float* w_proj   = (const float*)d_in[7];
  const float* b_proj   = (const float*)d_in[8];
  float* out = (float*)d_out;

  // workspace layout (floats)
  float* qkv   = (float*)d_ws;                                // B*3*C*T   (24 MB)
  float* S     = qkv   + (size_t)Bn * 3 * Cc * Tt;            // B*H*T*T   (128 MB)
  float* out1  = S     + (size_t)Bn * HEADS * Tt * Tt;        // B*T*C     (8 MB)
  float* sumsq = out1  + (size_t)Bn * Tt * Cc;                // B*H
  float* vsum  = sumsq + Bn * HEADS;                          // B*C

  hipMemsetAsync(sumsq, 0, Bn * HEADS * sizeof(float), stream);

  qkv_kernel    <<<dim3(Tt / 64, Cc / 16, Bn * 3),     32,  0, stream>>>(x, w_q, w_k, w_v, qkv);
  score_kernel  <<<dim3(Tt / 64, Tt / 16, Bn * HEADS), 32,  0, stream>>>(qkv, w_head, S);
  softmax_kernel<<<dim3(Tt, Bn * HEADS),               256, 0, stream>>>(S, sumsq);
  vsum_kernel   <<<dim3((Bn * Cc) / 256),              256, 0, stream>>>(qkv, vsum);
  av_kernel     <<<dim3(Tt / 16, Bn * HEADS),          32,  0, stream>>>(qkv, S, sumsq, vsum,
                                                                         in_gamma, in_beta, out1);
  proj_kernel   <<<dim3(Tt / 64, Cc / 16, Bn),         32,  0, stream>>>(out1, w_proj, b_proj, out);
}